// AGRNNCell_13211319403253
// MI455X (gfx1250) — compile-verified
//
#include <hip/hip_runtime.h>
#include <math.h>

typedef __attribute__((ext_vector_type(2))) float v2f;
typedef __attribute__((ext_vector_type(8))) float v8f;

static constexpr int Bb  = 8;
static constexpr int Nn  = 2000;
static constexpr int Din = 32;
static constexpr int Hd  = 64;
static constexpr int Dd  = 128;
static constexpr int TK  = 20;
static constexpr int BN  = Bb * Nn;   // 16000
static constexpr int SP  = 2064;      // score-strip LDS stride (2064 % 64 == 16 -> no bank conflicts)

// ---------------------------------------------------------------------------
// fp32 WMMA GEMM, templated: C[M x (NT*16)] = A[M x K] @ W[K x NT*16]
// grid = (M/16), block = 32. One wave computes a 16 x (NT*16) row strip so the
// A-fragment is loaded once per K-step and reused for NT WMMAs (K fully
// unrolled at compile time).
// A-frag (16x4 f32): lane l (m=l&15, h=l>>4): {A[m][kk+2h], A[m][kk+2h+1]}
// B-frag (4x16 f32): lane l (n=l&15, h=l>>4): {W[kk+2h][n], W[kk+2h+1][n]}
// C/D (16x16 f32):   VGPR r, lane l: row = r + 8*(l>>4), col = l&15
// ---------------------------------------------------------------------------
template <int NT, int K>
__global__ __launch_bounds__(32) void gemm_f32_wmma(
    const float* __restrict__ A, int lda,
    const float* __restrict__ W, int ldw,
    const float* __restrict__ bias,
    const float* __restrict__ add, int ldadd,
    float* __restrict__ C, int ldc)
{
    const int m0   = blockIdx.x * 16;
    const int lane = threadIdx.x & 31;
    const int mn   = lane & 15;
    const int half = lane >> 4;

    v8f acc[NT];
#pragma unroll
    for (int t = 0; t < NT; ++t) acc[t] = (v8f){0.f,0.f,0.f,0.f,0.f,0.f,0.f,0.f};

    const float* arow = A + (size_t)(m0 + mn) * lda;
#pragma unroll
    for (int kk = 0; kk < K; kk += 4) {
        const v2f av = *(const v2f*)(arow + kk + 2*half);
        const float* wr0 = W + (size_t)(kk + 2*half)     * ldw + mn;
        const float* wr1 = W + (size_t)(kk + 2*half + 1) * ldw + mn;
#pragma unroll
        for (int t = 0; t < NT; ++t) {
            v2f bv;
            bv.x = wr0[t * 16];
            bv.y = wr1[t * 16];
            acc[t] = __builtin_amdgcn_wmma_f32_16x16x4_f32(
                         false, av, false, bv, (short)0, acc[t], false, false);
        }
    }

#pragma unroll
    for (int t = 0; t < NT; ++t) {
        const int col = t * 16 + mn;
        const float bval = bias ? bias[col] : 0.0f;
#pragma unroll
        for (int r = 0; r < 8; ++r) {
            const int row = m0 + r + 8*half;
            float v = acc[t][r] + bval;
            if (add) v += add[(size_t)row * ldadd + col];
            C[(size_t)row * ldc + col] = v;
        }
    }
}

// ---------------------------------------------------------------------------
// Zero a float region (float4 stores, n divisible by 4)
// ---------------------------------------------------------------------------
__global__ void zero_f4(float4* __restrict__ p, size_t n4)
{
    size_t i = (size_t)blockIdx.x * blockDim.x + threadIdx.x;
    const size_t stride = (size_t)gridDim.x * blockDim.x;
    const float4 z = make_float4(0.f, 0.f, 0.f, 0.f);
    for (; i < n4; i += stride) p[i] = z;
}

// xs[:, 64:128] = state
__global__ void copy_state_to_xs(const float* __restrict__ st, float* __restrict__ xs)
{
    int i = blockIdx.x * 256 + threadIdx.x;
    if (i < BN * Hd) {
        int node = i >> 6, c = i & 63;
        xs[(size_t)node * Dd + Hd + c] = st[i];
    }
}

// ---------------------------------------------------------------------------
// Attention: per (batch, 16-row strip): scores strip 16x2000 in LDS via WMMA
// (q-fragments cached in registers across all j-tiles), top-20 per row
// (iterative wave argmax), masked softmax, scatter A^T, sparse ctx = attn @ v.
// grid = (125, 8), block = 256 (8 waves). LDS ~132 KB (CDNA5: 320 KB/WGP).
// ---------------------------------------------------------------------------
__global__ __launch_bounds__(256) void attn_topk_ctx(
    const float* __restrict__ q, const float* __restrict__ k,
    const float* __restrict__ v,
    float* __restrict__ attnv, int* __restrict__ topi,
    float* __restrict__ ctx, float* __restrict__ Aout)
{
    __shared__ float S[16 * SP];         // ~132 KB score strip
    __shared__ float s_av[16 * TK];
    __shared__ int   s_tj[16 * TK];

    const int b    = blockIdx.y;
    const int i0   = blockIdx.x * 16;
    const int base = b * Nn;
    const int tid  = threadIdx.x;
    const int wave = tid >> 5, lane = tid & 31;
    const int mn = lane & 15, half = lane >> 4;

    // Cache the 16 q A-fragments for this row strip in registers (32 VGPRs):
    // reused across every j-tile this wave processes.
    v2f qf[Hd / 4];
    {
        const float* qrow = q + (size_t)(base + i0 + mn) * Hd;
#pragma unroll
        for (int s = 0; s < Hd / 4; ++s)
            qf[s] = *(const v2f*)(qrow + 4 * s + 2 * half);
    }

    // Phase 1: score strip via fp32 WMMA (scores = q @ k^T / 8)
    for (int jt = wave; jt < Nn / 16; jt += 8) {
        const int j0 = jt * 16;
        v8f acc = {0.f,0.f,0.f,0.f,0.f,0.f,0.f,0.f};
        const float* krow = k + (size_t)(base + j0 + mn) * Hd;  // B: col = j
#pragma unroll
        for (int s = 0; s < Hd / 4; ++s) {
            const v2f bv = *(const v2f*)(krow + 4 * s + 2 * half);
            acc = __builtin_amdgcn_wmma_f32_16x16x4_f32(
                      false, qf[s], false, bv, (short)0, acc, false, false);
        }
#pragma unroll
        for (int r = 0; r < 8; ++r)
            S[(r + 8*half) * SP + j0 + mn] = acc[r] * 0.125f;
    }
    __syncthreads();

    // Phase 2: top-20 + softmax. Wave w owns rows w and w+8.
    for (int rr = 0; rr < 2; ++rr) {
        const int row = wave + 8 * rr;
        float tv[TK]; int tj[TK];
        for (int t = 0; t < TK; ++t) {
            float best = -3.0e38f; int bj = 0;
            for (int j = lane; j < Nn; j += 32) {
                float s = S[row * SP + j];
                if (s > best) { best = s; bj = j; }
            }
            for (int off = 16; off > 0; off >>= 1) {
                float ov = __shfl_xor(best, off, 32);
                int   oj = __shfl_xor(bj,   off, 32);
                if (ov > best || (ov == best && oj < bj)) { best = ov; bj = oj; }
            }
            tv[t] = best; tj[t] = bj;
            if (lane == 0) S[row * SP + bj] = -3.0e38f;  // same-wave LDS: in order
        }
        // masked softmax: everything outside top-20 underflows to exactly 0
        const float mx = tv[0];
        float Z = 0.f;
        for (int t = 0; t < TK; ++t) { tv[t] = expf(tv[t] - mx); Z += tv[t]; }
        const float iZ = 1.0f / Z;
        if (lane < TK) {
            const float a = tv[lane] * iZ;
            const int gi  = base + i0 + row;
            attnv[(size_t)gi * TK + lane] = a;
            topi [(size_t)gi * TK + lane] = tj[lane];
            s_av[row * TK + lane] = a;
            s_tj[row * TK + lane] = tj[lane];
            // A = attn^T : A[b, j, i] = attn[b, i, j]
            Aout[((size_t)b * Nn + tj[lane]) * Nn + (i0 + row)] = a;
        }
    }
    __syncthreads();

    // Phase 3: ctx[i,:] = sum_t attn[i,t] * v[topi[i,t], :]
    const int row = tid >> 4, c = tid & 15;
    float a0 = 0.f, a1 = 0.f, a2 = 0.f, a3 = 0.f;
    for (int t = 0; t < TK; ++t) {
        const int   j = s_tj[row * TK + t];
        const float w = s_av[row * TK + t];
        const float* vp = v + (size_t)(base + j) * Hd;
        if (t + 1 < TK)   // prefetch next gathered row (global_prefetch_b8)
            __builtin_prefetch(v + (size_t)(base + s_tj[row * TK + t + 1]) * Hd, 0, 1);
        a0 += w * vp[c];      a1 += w * vp[c + 16];
        a2 += w * vp[c + 32]; a3 += w * vp[c + 48];
    }
    float* cp = ctx + (size_t)(base + i0 + row) * Hd;
    cp[c] = a0; cp[c + 16] = a1; cp[c + 32] = a2; cp[c + 48] = a3;
}

// ---------------------------------------------------------------------------
// LayerNorm over D=128, one wave per row. grid = BN/8, block = 256.
// ---------------------------------------------------------------------------
__global__ __launch_bounds__(256) void layernorm128(
    const float* __restrict__ xp, const float* __restrict__ g,
    const float* __restrict__ bt, float* __restrict__ xo)
{
    const int row  = blockIdx.x * 8 + (threadIdx.x >> 5);
    const int lane = threadIdx.x & 31;
    const float* xr = xp + (size_t)row * Dd;
    const int c0 = lane, c1 = lane + 32, c2 = lane + 64, c3 = lane + 96;
    float v0 = xr[c0], v1 = xr[c1], v2 = xr[c2], v3 = xr[c3];
    float s = v0 + v1 + v2 + v3;
    for (int off = 16; off > 0; off >>= 1) s += __shfl_xor(s, off, 32);
    const float mu = s * (1.0f / Dd);
    const float d0 = v0 - mu, d1 = v1 - mu, d2 = v2 - mu, d3 = v3 - mu;
    float sq = d0*d0 + d1*d1 + d2*d2 + d3*d3;
    for (int off = 16; off > 0; off >>= 1) sq += __shfl_xor(sq, off, 32);
    const float inv = rsqrtf(sq * (1.0f / Dd) + 1e-6f);
    float* xw = xo + (size_t)row * Dd;
    xw[c0] = d0 * inv * g[c0] + bt[c0];
    xw[c1] = d1 * inv * g[c1] + bt[c1];
    xw[c2] = d2 * inv * g[c2] + bt[c2];
    xw[c3] = d3 * inv * g[c3] + bt[c3];
}

// deg = 1 (self loop) + sum of attn row mass captured by top-20 edges
__global__ void compute_dinv(const float* __restrict__ attnv, float* __restrict__ dinv)
{
    int i = blockIdx.x * 256 + threadIdx.x;
    if (i < BN) {
        float d = 1.0f;
        for (int t = 0; t < TK; ++t) d += attnv[(size_t)i * TK + t];
        dinv[i] = d > 0.f ? rsqrtf(d) : 0.f;
    }
}

// ---------------------------------------------------------------------------
// GCN aggregation for z and r gates; also produce z*state.
// out = dinv[i]*(sum_t ew*dinv[row_t]*H[row_t]) + dinv[i]^2*H[i] + bias
// block 256 -> 4 nodes, 64 feature lanes each.
// ---------------------------------------------------------------------------
__global__ __launch_bounds__(256) void agg_zr(
    const float* __restrict__ Hg1, const float* __restrict__ Hg2,
    const float* __restrict__ g1b, const float* __restrict__ g2b,
    const float* __restrict__ dinv, const int* __restrict__ topi,
    const float* __restrict__ attnv, const float* __restrict__ st,
    float* __restrict__ rbuf, float* __restrict__ zst)
{
    const int node = blockIdx.x * 4 + (threadIdx.x >> 6);
    const int c    = threadIdx.x & 63;
    const float di = dinv[node];
    float a1 = 0.f, a2 = 0.f;
    for (int t = 0; t < TK; ++t) {
        const int   rn = topi[(size_t)node * TK + t];
        const float w  = attnv[(size_t)node * TK + t] * dinv[rn];
        if (t + 1 < TK) {
            const int nx = topi[(size_t)node * TK + t + 1];
            __builtin_prefetch(Hg1 + (size_t)nx * Hd, 0, 1);
            __builtin_prefetch(Hg2 + (size_t)nx * Hd, 0, 1);
        }
        a1 += w * Hg1[(size_t)rn * Hd + c];
        a2 += w * Hg2[(size_t)rn * Hd + c];
    }
    const size_t idx = (size_t)node * Hd + c;
    const float o1 = di * a1 + di * di * Hg1[idx] + g1b[c];
    const float o2 = di * a2 + di * di * Hg2[idx] + g2b[c];
    const float z = 1.0f / (1.0f + expf(-o1));
    const float r = 1.0f / (1.0f + expf(-o2));
    rbuf[idx] = r;
    zst[idx]  = z * st[idx];
}

// Final aggregation: hc = tanh(agg(Hup)+up_b); h = r*st + (1-r)*hc
__global__ __launch_bounds__(256) void agg_h(
    const float* __restrict__ Hup, const float* __restrict__ upb,
    const float* __restrict__ dinv, const int* __restrict__ topi,
    const float* __restrict__ attnv, const float* __restrict__ st,
    const float* __restrict__ rbuf, float* __restrict__ hout)
{
    const int node = blockIdx.x * 4 + (threadIdx.x >> 6);
    const int c    = threadIdx.x & 63;
    const float di = dinv[node];
    float a = 0.f;
    for (int t = 0; t < TK; ++t) {
        const int   rn = topi[(size_t)node * TK + t];
        const float w  = attnv[(size_t)node * TK + t] * dinv[rn];
        if (t + 1 < TK)
            __builtin_prefetch(Hup + (size_t)topi[(size_t)node * TK + t + 1] * Hd, 0, 1);
        a += w * Hup[(size_t)rn * Hd + c];
    }
    const size_t idx = (size_t)node * Hd + c;
    const float o  = di * a + di * di * Hup[idx] + upb[c];
    const float hc = tanhf(o);
    const float r  = rbuf[idx];
    hout[idx] = r * st[idx] + (1.0f - r) * hc;
}

// ---------------------------------------------------------------------------
extern "C" void kernel_launch(void* const* d_in, const int* in_sizes, int n_in,
                              void* d_out, int out_size, void* d_ws, size_t ws_size,
                              hipStream_t stream)
{
    const float* x     = (const float*)d_in[0];
    const float* state = (const float*)d_in[1];
    const float* lin_W = (const float*)d_in[2];
    const float* lin_b = (const float*)d_in[3];
    const float* Wq    = (const float*)d_in[4];
    const float* bq    = (const float*)d_in[5];
    const float* Wk    = (const float*)d_in[6];
    const float* bk    = (const float*)d_in[7];
    const float* Wv    = (const float*)d_in[8];
    const float* bv    = (const float*)d_in[9];
    const float* Wo    = (const float*)d_in[10];
    const float* bo    = (const float*)d_in[11];
    const float* ln_g  = (const float*)d_in[12];
    const float* ln_b  = (const float*)d_in[13];
    const float* g1_W  = (const float*)d_in[14];
    const float* g1_b  = (const float*)d_in[15];
    const float* g2_W  = (const float*)d_in[16];
    const float* g2_b  = (const float*)d_in[17];
    const float* up_W  = (const float*)d_in[18];
    const float* up_b  = (const float*)d_in[19];
    // d_in[20] = top_k (fixed at 20)

    float* out  = (float*)d_out;
    float* hout = out;                         // (B,N,Hd) = 1,024,000 floats
    float* Aout = out + (size_t)BN * Hd;       // (B,N,N)  = 32,000,000 floats

    // workspace layout (floats)
    float* ws    = (float*)d_ws;
    float* xs    = ws;                              // BN*128
    float* qb    = xs    + (size_t)BN * Dd;         // BN*64
    float* kb    = qb    + (size_t)BN * Hd;         // BN*64
    float* vb    = kb    + (size_t)BN * Hd;         // BN*64
    float* attnv = vb    + (size_t)BN * Hd;         // BN*20
    int*   topi  = (int*)(attnv + (size_t)BN * TK); // BN*20
    float* ctx   = attnv + (size_t)2 * BN * TK;     // BN*64
    float* xxp   = ctx   + (size_t)BN * Hd;         // BN*128
    float* xx    = xxp   + (size_t)BN * Dd;         // BN*128
    float* dinv  = xx    + (size_t)BN * Dd;         // BN
    float* rbuf  = dinv  + (size_t)BN;              // BN*64
    float* zst   = rbuf  + (size_t)BN * Hd;         // BN*64
    float* Hg1   = qb;   // reuse q/k/v after attention
    float* Hg2   = kb;
    float* Hup   = vb;

    const int Mt = BN / 16;   // 1000 row tiles

    // 0) zero the A output region (attn^T is sparse: only top-20 scattered)
    zero_f4<<<8192, 256, 0, stream>>>((float4*)Aout, (size_t)BN * Nn / 4);

    // 1) xs[:, :64] = x @ lin_W + lin_b ; xs[:, 64:] = state
    gemm_f32_wmma<4, Din><<<Mt, 32, 0, stream>>>(
        x, Din, lin_W, Hd, lin_b, nullptr, 0, xs, Dd);
    copy_state_to_xs<<<(BN * Hd + 255) / 256, 256, 0, stream>>>(state, xs);

    // 2) q, k, v
    gemm_f32_wmma<4, Dd><<<Mt, 32, 0, stream>>>(xs, Dd, Wq, Hd, bq, nullptr, 0, qb, Hd);
    gemm_f32_wmma<4, Dd><<<Mt, 32, 0, stream>>>(xs, Dd, Wk, Hd, bk, nullptr, 0, kb, Hd);
    gemm_f32_wmma<4, Dd><<<Mt, 32, 0, stream>>>(xs, Dd, Wv, Hd, bv, nullptr, 0, vb, Hd);

    // 3) attention: scores (WMMA) -> top-20 -> softmax -> A^T scatter + ctx
    attn_topk_ctx<<<dim3(Nn/16, Bb), 256, 0, stream>>>(qb, kb, vb, attnv, topi, ctx, Aout);

    // 4) xx = LN(ctx @ Wo + bo + xs) * ln_g + ln_b
    gemm_f32_wmma<8, Hd><<<Mt, 32, 0, stream>>>(ctx, Hd, Wo, Dd, bo, xs, Dd, xxp, Dd);
    layernorm128<<<BN / 8, 256, 0, stream>>>(xxp, ln_g, ln_b, xx);

    // 5) GCN feature transforms (biases added post-aggregation, per reference)
    gemm_f32_wmma<4, Dd><<<Mt, 32, 0, stream>>>(xx, Dd, g1_W, Hd, nullptr, nullptr, 0, Hg1, Hd);
    gemm_f32_wmma<4, Dd><<<Mt, 32, 0, stream>>>(xx, Dd, g2_W, Hd, nullptr, nullptr, 0, Hg2, Hd);
    gemm_f32_wmma<4, Dd><<<Mt, 32, 0, stream>>>(xx, Dd, up_W, Hd, nullptr, nullptr, 0, Hup, Hd);

    // 6) degree^{-1/2}
    compute_dinv<<<(BN + 255) / 256, 256, 0, stream>>>(attnv, dinv);

    // 7) z, r gates via sparse aggregation; zst = z * state
    agg_zr<<<BN / 4, 256, 0, stream>>>(Hg1, Hg2, g1_b, g2_b, dinv, topi, attnv,
                                       state, rbuf, zst);

    // 8) Hup += (z*state) @ up_W[128:192, :]
    gemm_f32_wmma<4, Hd><<<Mt, 32, 0, stream>>>(
        zst, Hd, up_W + (size_t)Dd * Hd, Hd, nullptr, Hup, Hd, Hup, Hd);

    // 9) hc = tanh(agg(Hup) + up_b); h = r*st + (1-r)*hc
    agg_h<<<BN / 4, 256, 0, stream>>>(Hup, up_b, dinv, topi, attnv, state, rbuf, hout);
}